// DigitCapsLayer_1340029796593
// MI455X (gfx1250) — compile-verified
//
#include <hip/hip_runtime.h>
#include <math.h>

#define B_    256
#define INU   8
#define C_    1152
#define U_    10
#define OUT_  16
#define UHAT_ELEMS ((size_t)B_ * C_ * U_ * OUT_)   // 47,185,920 floats = 189 MB

typedef __attribute__((ext_vector_type(2))) float v2f;
typedef __attribute__((ext_vector_type(8))) float v8f;

// ---------------------------------------------------------------------------
// Zero the routing logits b[c,u]
// ---------------------------------------------------------------------------
__global__ void zero_kernel(float* __restrict__ p, int n) {
    int i = blockIdx.x * blockDim.x + threadIdx.x;
    if (i < n) p[i] = 0.0f;
}

// ---------------------------------------------------------------------------
// u_hat[c][u][b][o] = sum_i x[b][i][c] * W[c][u][o][i]
// One wave (32 lanes) per (c, batch-tile of 16). Per u (N-tile of 16 = one u,
// since OUT_==16): two V_WMMA_F32_16X16X4_F32 covering K=0..3 and K=4..7.
//
// fp32 16x4 A layout: lanes 0-15 hold M=0..15 / K={0,1} in VGPR{0,1};
//                     lanes 16-31 hold M=0..15 / K={2,3}.
// B (4x16) mirrors with N striped across lanes, K across half+vgpr.
// C/D (16x16): VGPR r, lane l -> row M = r + 8*(l>>4), col N = l&15.
// ---------------------------------------------------------------------------
__global__ __launch_bounds__(32)
void uhat_kernel(const float* __restrict__ x, const float* __restrict__ W,
                 float* __restrict__ uhat) {
    const int c    = blockIdx.x;          // capsule
    const int bt   = blockIdx.y;          // batch tile (16 tiles of 16)
    const int lane = threadIdx.x;
    const int m    = lane & 15;           // row within tile / column o
    const int h    = lane >> 4;           // lane half
    const int kb   = h * 2;               // K base for this half (0 or 2)

    // A fragment: x[b][i][c], b = bt*16+m, i = K
    const float* xb = x + (size_t)(bt * 16 + m) * (INU * C_) + c;
    v2f A0, A1;
    A0[0] = xb[(size_t)(kb    ) * C_];
    A0[1] = xb[(size_t)(kb + 1) * C_];
    A1[0] = xb[(size_t)(kb + 4) * C_];
    A1[1] = xb[(size_t)(kb + 5) * C_];

    for (int u = 0; u < U_; ++u) {
        // B fragment: W[c][u][o][i], o = m (column), i = K
        const float* wb = W + ((size_t)(c * U_ + u) * OUT_ + m) * INU;
        v2f B0, B1;
        B0[0] = wb[kb];     B0[1] = wb[kb + 1];
        B1[0] = wb[kb + 4]; B1[1] = wb[kb + 5];

        v8f acc = {0.f, 0.f, 0.f, 0.f, 0.f, 0.f, 0.f, 0.f};
        acc = __builtin_amdgcn_wmma_f32_16x16x4_f32(
                  false, A0, false, B0, (short)0, acc, false, false);
        acc = __builtin_amdgcn_wmma_f32_16x16x4_f32(
                  false, A1, false, B1, (short)0, acc, false, false);

        // Store D tile into u_hat[c][u][bt*16 + M][o]
        float* dst = uhat + ((size_t)(c * U_ + u) * B_ + bt * 16) * OUT_;
        #pragma unroll
        for (int r = 0; r < 8; ++r)
            dst[(size_t)(r + 8 * h) * OUT_ + m] = acc[r];
    }
}

// ---------------------------------------------------------------------------
// cc[c][u] = softmax over u of bb[c][u]   (one thread per c, 10 elements)
// ---------------------------------------------------------------------------
__global__ void softmax_kernel(const float* __restrict__ bb,
                               float* __restrict__ cc) {
    int c = blockIdx.x * blockDim.x + threadIdx.x;
    if (c >= C_) return;
    float e[U_];
    float mx = -3.402823e38f;
    #pragma unroll
    for (int u = 0; u < U_; ++u) mx = fmaxf(mx, bb[c * U_ + u]);
    float s = 0.f;
    #pragma unroll
    for (int u = 0; u < U_; ++u) { float t = expf(bb[c * U_ + u] - mx); e[u] = t; s += t; }
    float inv = 1.f / s;
    #pragma unroll
    for (int u = 0; u < U_; ++u) cc[c * U_ + u] = e[u] * inv;
}

// ---------------------------------------------------------------------------
// s[b][u][o] = sum_c cc[c][u] * u_hat[c][u][b][o]; then squash over o.
// Grid: (u, 16 chunks of 256 (b,o) pairs). o lives in the low 4 lane bits, so
// the norm reduction over o is 4 xor-shuffles inside a 16-lane group (wave32).
// ---------------------------------------------------------------------------
__global__ __launch_bounds__(256)
void s_kernel(const float* __restrict__ uhat, const float* __restrict__ cc,
              float* __restrict__ vout) {
    const int u = blockIdx.x;
    const int j = blockIdx.y * 256 + threadIdx.x;   // 0..4095 = b*16+o
    const int b = j >> 4;
    const int o = j & 15;

    __shared__ float ccs[C_];
    for (int i = threadIdx.x; i < C_; i += 256) ccs[i] = cc[i * U_ + u];
    __syncthreads();

    const float* uh = uhat + (size_t)(u * B_ + b) * OUT_ + o;  // + c*40960
    float s = 0.f;
    for (int c = 0; c < C_; ++c)
        s = fmaf(ccs[c], uh[(size_t)c * (U_ * B_ * OUT_)], s);

    float n2 = s * s;
    n2 += __shfl_xor(n2, 1);
    n2 += __shfl_xor(n2, 2);
    n2 += __shfl_xor(n2, 4);
    n2 += __shfl_xor(n2, 8);

    float v = s * sqrtf(n2) / (1.f + n2);   // n2/(1+n2) * s/sqrt(n2)
    vout[(size_t)(b * U_ + u) * OUT_ + o] = v;
}

// ---------------------------------------------------------------------------
// bb[c][u] += (1/B) * sum_{b,o} u_hat[c][u][b][o] * v[b][u][o]
// One wave per (c,u): 4096 contiguous u_hat floats reduced across 32 lanes.
// ---------------------------------------------------------------------------
__global__ __launch_bounds__(256)
void agree_kernel(const float* __restrict__ uhat, const float* __restrict__ v,
                  float* __restrict__ bb) {
    const int wid  = threadIdx.x >> 5;
    const int lane = threadIdx.x & 31;
    const int cu   = blockIdx.x * 8 + wid;   // 1440*8 = 11520 exactly
    const int u    = cu % U_;

    const float* uh = uhat + (size_t)cu * (B_ * OUT_);
    float acc = 0.f;
    for (int i = lane; i < B_ * OUT_; i += 32) {
        int b = i >> 4, o = i & 15;
        acc = fmaf(uh[i], v[(size_t)(b * U_ + u) * OUT_ + o], acc);
    }
    acc += __shfl_xor(acc, 1);
    acc += __shfl_xor(acc, 2);
    acc += __shfl_xor(acc, 4);
    acc += __shfl_xor(acc, 8);
    acc += __shfl_xor(acc, 16);
    if (lane == 0) bb[cu] += acc * (1.0f / B_);
}

// ---------------------------------------------------------------------------
extern "C" void kernel_launch(void* const* d_in, const int* in_sizes, int n_in,
                              void* d_out, int out_size, void* d_ws, size_t ws_size,
                              hipStream_t stream) {
    (void)in_sizes; (void)n_in; (void)out_size; (void)ws_size;

    const float* x = (const float*)d_in[0];   // [256, 8, 1152]
    const float* W = (const float*)d_in[1];   // [1152, 10, 16, 8]

    float* uhat = (float*)d_ws;               // 189 MB, L2-resident
    float* bb   = uhat + UHAT_ELEMS;          // routing logits [C,U]
    float* cc   = bb + (size_t)C_ * U_;       // softmax coeffs [C,U]
    float* vv   = cc + (size_t)C_ * U_;       // v [B,U,OUT]

    zero_kernel<<<(C_ * U_ + 255) / 256, 256, 0, stream>>>(bb, C_ * U_);
    uhat_kernel<<<dim3(C_, B_ / 16), 32, 0, stream>>>(x, W, uhat);

    for (int it = 0; it < 4; ++it) {
        softmax_kernel<<<(C_ + 255) / 256, 256, 0, stream>>>(bb, cc);
        float* vout = (it == 3) ? (float*)d_out : vv;
        s_kernel<<<dim3(U_, (B_ * OUT_) / 256), 256, 0, stream>>>(uhat, cc, vout);
        if (it < 3)  // last iteration's logit update is dead code
            agree_kernel<<<(C_ * U_) / 8, 256, 0, stream>>>(uhat, vv, bb);
    }
}